// HeatmapMaxDetBlock_20504173871276
// MI455X (gfx1250) — compile-verified
//
#include <hip/hip_runtime.h>
#include <stdint.h>

// Problem constants from the reference: x is [B=64, C=17, H=256, W=192] f32.
#define HM_H 256
#define HM_W 192
#define HM_HW (HM_H * HM_W)           // 49152 floats per row
#define NTHREADS 256                  // 8 wave32s per block
#define V4_PER_ROW (HM_HW / 4)        // 12288 float4 per row
#define ITERS (V4_PER_ROW / NTHREADS) // 48 float4 per thread
#define BATCH 8                       // float4 per pipeline stage per thread
#define NSTAGES (ITERS / BATCH)       // 6 stages
#define PF_DIST (2 * BATCH * NTHREADS) // prefetch 2 stages (64 KB) ahead

// Monotonic mapping float -> uint32 so that unsigned compare == float compare.
__device__ __forceinline__ uint32_t ordered_bits(float f) {
    uint32_t u = __float_as_uint(f);
    return (u & 0x80000000u) ? ~u : (u | 0x80000000u);
}
__device__ __forceinline__ float unordered_bits(uint32_t o) {
    uint32_t u = (o & 0x80000000u) ? (o ^ 0x80000000u) : ~o;
    return __uint_as_float(u);
}

// Branch-free running-argmax update (lowers to v_max/v_cmp/v_cndmask only —
// no exec-mask branches, so the scheduler can keep loads in flight).
__device__ __forceinline__ void upd(float4 v, int e0, float& best, int& bidx) {
    const float m4 = fmaxf(fmaxf(v.x, v.y), fmaxf(v.z, v.w));
    // first index among the 4 equal to m4 (first-occurrence tie break)
    const int off = (v.x == m4) ? 0 : (v.y == m4) ? 1 : (v.z == m4) ? 2 : 3;
    const int cand = e0 + off;
    const bool gt = (m4 > best);  // strict >: keep earlier index on ties
    bidx = gt ? cand : bidx;
    best = gt ? m4 : best;
}

// One block per (b, c) row: streaming argmax over 49152 floats, then the
// O(1) quarter-pixel refinement done by thread 0.
__global__ __launch_bounds__(NTHREADS) void heatmap_maxdet_kernel(
    const float* __restrict__ x, float* __restrict__ out) {
    const int row = blockIdx.x;
    const float4* __restrict__ rowp =
        reinterpret_cast<const float4*>(x) + (size_t)row * V4_PER_ROW;
    const int tid = threadIdx.x;

    float best = -__builtin_inff();
    int bidx = 0;

    // Double-buffered register pipeline: 8 outstanding global_load_b128 per
    // wave while the previous batch is reduced (branch-free).
    float4 buf[2][BATCH];

#pragma unroll
    for (int j = 0; j < BATCH; ++j)
        buf[0][j] = rowp[j * NTHREADS + tid];

#pragma unroll
    for (int s = 0; s < NSTAGES; ++s) {
        const int cb = s & 1;
        if (s + 1 < NSTAGES) {
            const int nb = cb ^ 1;
#pragma unroll
            for (int j = 0; j < BATCH; ++j) {
                const int v4i = ((s + 1) * BATCH + j) * NTHREADS + tid;
                // gfx1250 speculative prefetch (global_prefetch_b8, L2 only);
                // OOB prefetch is silently dropped per ISA 10.5.
                __builtin_prefetch((const void*)(rowp + v4i + PF_DIST), 0, 1);
                buf[nb][j] = rowp[v4i];
            }
        }
#pragma unroll
        for (int j = 0; j < BATCH; ++j) {
            const int e0 = ((s * BATCH + j) * NTHREADS + tid) * 4;
            upd(buf[cb][j], e0, best, bidx);
        }
    }

    // Pack (value, first-index) into one u64 key: max over keys == argmax with
    // jnp-style first-occurrence tie break (~idx: larger key <=> smaller idx).
    uint64_t key = ((uint64_t)ordered_bits(best) << 32) |
                   (uint32_t)(~(uint32_t)bidx);

    // wave32 reduction (5 xor-shuffle steps), then combine the 8 wave leaders.
    for (int m = 16; m > 0; m >>= 1) {
        uint64_t other = (uint64_t)__shfl_xor((unsigned long long)key, m, 32);
        key = (other > key) ? other : key;
    }

    __shared__ uint64_t wkey[NTHREADS / 32];
    const int wave = tid >> 5;
    const int lane = tid & 31;
    if (lane == 0) wkey[wave] = key;
    __syncthreads();

    if (tid == 0) {
        uint64_t k = wkey[0];
#pragma unroll
        for (int w = 1; w < NTHREADS / 32; ++w)
            k = (wkey[w] > k) ? wkey[w] : k;

        const float score = unordered_bits((uint32_t)(k >> 32)); // bit-exact max
        const int idx = (int)(~(uint32_t)k);                     // argmax (first)

        const bool msk = (score > 0.0f);
        const int qx = idx % HM_W;
        const int qy = idx / HM_W;
        const int pxi = msk ? qx : 0;
        const int pyi = msk ? qy : 0;
        float px = (float)pxi;
        float py = (float)pyi;

        const bool interior =
            (pxi > 0) && (pxi < HM_W - 1) && (pyi > 0) && (pyi < HM_H - 1);
        if (interior) {
            const float* rp = x + (size_t)row * HM_HW;
            const int base = pyi * HM_W + pxi;  // == idx when masked-interior
            const float r = rp[base + 1];
            const float l = rp[base - 1];
            const float u = rp[base + HM_W];
            const float d = rp[base - HM_W];
            // sign(diff) * 0.25  (sign(0) == 0, matching jnp.sign)
            px += (r > l) ? 0.25f : ((r < l) ? -0.25f : 0.0f);
            py += (u > d) ? 0.25f : ((u < d) ? -0.25f : 0.0f);
        }

        out[row * 3 + 0] = px;
        out[row * 3 + 1] = py;
        out[row * 3 + 2] = score;
    }
}

extern "C" void kernel_launch(void* const* d_in, const int* in_sizes, int n_in,
                              void* d_out, int out_size, void* d_ws,
                              size_t ws_size, hipStream_t stream) {
    const float* x = (const float*)d_in[0];
    float* out = (float*)d_out;
    const int rows = in_sizes[0] / HM_HW;  // B*C = 1088
    heatmap_maxdet_kernel<<<dim3(rows), dim3(NTHREADS), 0, stream>>>(x, out);
}